// NodeClassifier_81810537054299
// MI455X (gfx1250) — compile-verified
//
#include <hip/hip_runtime.h>

typedef __attribute__((ext_vector_type(2))) float v2f;
typedef __attribute__((ext_vector_type(8))) float v8f;

// ---------------------------------------------------------------------------
// GEMM + bias:  H[m,n] = sum_k relu?(A[m,k]) * W[k,n] + bias[n]
// A: [M,K] row-major, W: [K,DOUT] row-major, H: [M,DOUT] row-major.
// One wave per 16x16 tile via V_WMMA_F32_16X16X4_F32. K and DOUT are
// compile-time so every address is a 32-bit add; NO guards anywhere
// (ragged layer-3 handled by zero-padding W/bias/H to DOUT=48).
// ---------------------------------------------------------------------------
template <bool RELU_A, int K, int DOUT>
__global__ __launch_bounds__(32) void wmma_gemm_bias_kernel(
    const float* __restrict__ A, const float* __restrict__ W,
    const float* __restrict__ bias, float* __restrict__ H) {
  const int lane = threadIdx.x;   // 0..31, full wave, EXEC all ones
  const int half = lane >> 4;     // 0: K pair {0,1}/rows 0-7 ; 1: {2,3}/rows 8-15
  const int l15  = lane & 15;
  const int m0 = blockIdx.x << 4;
  const int n0 = blockIdx.y << 4;
  const int ncol = n0 + l15;

  // A operand: lane holds row (m0+l15), K elements (k + 2*half, k + 2*half + 1)
  const float* arow = A + (m0 + l15) * K + (half << 1);
  // B operand: lane holds column ncol, rows k + 2*half (+1)
  const float* wptr = W + (half << 1) * DOUT + ncol;

  v8f acc0 = {};
  v8f acc1 = {};

#pragma unroll 4
  for (int k = 0; k < K; k += 8) {
    float2 av0 = *(const float2*)(arow);
    float2 av1 = *(const float2*)(arow + 4);
    arow += 8;
    float bx0 = wptr[0];
    float by0 = wptr[DOUT];
    float bx1 = wptr[4 * DOUT];
    float by1 = wptr[5 * DOUT];
    wptr += 8 * DOUT;
    if (RELU_A) {
      av0.x = fmaxf(av0.x, 0.0f);
      av0.y = fmaxf(av0.y, 0.0f);
      av1.x = fmaxf(av1.x, 0.0f);
      av1.y = fmaxf(av1.y, 0.0f);
    }
    v2f a0; a0.x = av0.x; a0.y = av0.y;
    v2f a1; a1.x = av1.x; a1.y = av1.y;
    v2f b0; b0.x = bx0;   b0.y = by0;
    v2f b1; b1.x = bx1;   b1.y = by1;

    acc0 = __builtin_amdgcn_wmma_f32_16x16x4_f32(false, a0, false, b0,
                                                 (short)0, acc0, false, false);
    acc1 = __builtin_amdgcn_wmma_f32_16x16x4_f32(false, a1, false, b1,
                                                 (short)0, acc1, false, false);
  }

  const float bv = bias[ncol];
  // C/D layout: VGPR r holds M = r + 8*half, N = l15
  float* orow = H + (m0 + (half << 3)) * DOUT + ncol;
#pragma unroll
  for (int r = 0; r < 8; ++r) {
    orow[r * DOUT] = acc0[r] + acc1[r] + bv;
  }
}

// ---------------------------------------------------------------------------
// Scatter-sum over edges: out[dst[e], c] += H[src[e], c], c in [0,D).
// H has row stride LD (padded for layer 3). One thread per (edge, channel);
// for D=256 a block covers exactly one edge (uniform src/dst, coalesced
// 1KB load + contiguous f32 atomics resolved in the 192MB L2).
// ---------------------------------------------------------------------------
template <int D, int LD>
__global__ void scatter_add_kernel(const float* __restrict__ H,
                                   const int* __restrict__ src,
                                   const int* __restrict__ dst,
                                   float* __restrict__ out, int E) {
  long long idx = (long long)blockIdx.x * blockDim.x + threadIdx.x;
  if (idx >= (long long)E * D) return;
  int e = (int)(idx / D);
  int c = (int)(idx - (long long)e * D);
  int s = src[e];
  int t = dst[e];
  atomicAdd(out + (size_t)t * D + c, H[(size_t)s * LD + c]);
}

__global__ void zero_kernel(float* __restrict__ p, int n) {
  int idx = blockIdx.x * blockDim.x + threadIdx.x;
  if (idx < n) p[idx] = 0.0f;
}

__global__ void relu_kernel(float* __restrict__ p, int n) {
  int idx = blockIdx.x * blockDim.x + threadIdx.x;
  if (idx < n) p[idx] = fmaxf(p[idx], 0.0f);
}

// Zero-pad W3 [256,40] -> Wp [256,48] and b3 [40] -> bp [48].
__global__ void pad_w3_kernel(const float* __restrict__ W3,
                              const float* __restrict__ b3,
                              float* __restrict__ Wp, float* __restrict__ bp) {
  int idx = blockIdx.x * blockDim.x + threadIdx.x;
  if (idx < 256 * 48) {
    int r = idx / 48;
    int c = idx - r * 48;
    Wp[idx] = (c < 40) ? W3[r * 40 + c] : 0.0f;
  }
  if (idx < 48) bp[idx] = (idx < 40) ? b3[idx] : 0.0f;
}

// ---------------------------------------------------------------------------
extern "C" void kernel_launch(void* const* d_in, const int* in_sizes, int n_in,
                              void* d_out, int out_size, void* d_ws,
                              size_t ws_size, hipStream_t stream) {
  (void)in_sizes; (void)n_in; (void)out_size; (void)ws_size;

  const float* x  = (const float*)d_in[0];   // [10000, 256]
  const int*   ei = (const int*)d_in[1];     // [2, 320000] int32
  const float* W1 = (const float*)d_in[2];   // [256, 256]
  const float* b1 = (const float*)d_in[3];
  const float* W2 = (const float*)d_in[4];   // [256, 256]
  const float* b2 = (const float*)d_in[5];
  const float* W3 = (const float*)d_in[6];   // [256, 40]
  const float* b3 = (const float*)d_in[7];
  float* out = (float*)d_out;                // [10000, 40]

  constexpr int N = 10000, E = 320000, DH = 256, DC = 40, DCP = 48;
  const int* src = ei;
  const int* dst = ei + E;

  float* ws0 = (float*)d_ws;                 // N*256 floats
  float* ws1 = ws0 + (size_t)N * DH;         // N*256 floats
  float* wp  = ws1 + (size_t)N * DH;         // 256*48 floats (padded W3)
  float* bp  = wp + 256 * DCP;               // 48 floats (padded b3)

  const dim3 wave(32, 1, 1);
  const dim3 gemm_grid(N / 16, DH / 16);     // 625 x 16
  const dim3 gemm_grid3(N / 16, DCP / 16);   // 625 x 3

  const int nDH = N * DH;                    // 2,560,000
  const int nDC = N * DC;                    // 400,000
  const int zb = (nDH + 255) / 256;
  const int zc = (nDC + 255) / 256;
  const int sb256 = (int)(((long long)E * DH + 255) / 256);  // 320,000 blocks
  const int sb40  = (int)(((long long)E * DC + 255) / 256);  // 50,000 blocks

  // One-shot pad of W3/b3 into workspace (deterministic, re-done every call).
  pad_w3_kernel<<<48, 256, 0, stream>>>(W3, b3, wp, bp);

  // ---- Layer 1: H1 = x@W1 + b1 ; X2 = scatter(H1) ----
  wmma_gemm_bias_kernel<false, 256, 256><<<gemm_grid, wave, 0, stream>>>(x, W1, b1, ws0);
  zero_kernel<<<zb, 256, 0, stream>>>(ws1, nDH);
  scatter_add_kernel<DH, DH><<<sb256, 256, 0, stream>>>(ws0, src, dst, ws1, E);

  // ---- Layer 2: H2 = relu(X2)@W2 + b2 ; X3 = scatter(H2) ----
  wmma_gemm_bias_kernel<true, 256, 256><<<gemm_grid, wave, 0, stream>>>(ws1, W2, b2, ws0);
  zero_kernel<<<zb, 256, 0, stream>>>(ws1, nDH);
  scatter_add_kernel<DH, DH><<<sb256, 256, 0, stream>>>(ws0, src, dst, ws1, E);

  // ---- Layer 3 (padded to 48 cols): H3 = relu(X3)@Wp + bp ----
  wmma_gemm_bias_kernel<true, 256, DCP><<<gemm_grid3, wave, 0, stream>>>(ws1, wp, bp, ws0);
  zero_kernel<<<zc, 256, 0, stream>>>(out, nDC);
  scatter_add_kernel<DC, DCP><<<sb40, 256, 0, stream>>>(ws0, src, dst, out, E);
  relu_kernel<<<zc, 256, 0, stream>>>(out, nDC);
}